// Dilation2D_26860725469244
// MI455X (gfx1250) — compile-verified
//
#include <hip/hip_runtime.h>
#include <stdint.h>

// Problem constants (match reference module config)
#define N_    8
#define CIN_  4
#define H_    512
#define W_    512
#define COUT_ 16
#define KH_   5
#define KW_   5
#define OHA   2
#define OWA   2
#define PADV  (-1e30f)
#define NTAPS (CIN_ * KH_ * KW_)   // 100

// Tiling
#define TW 64
#define TH 8
#define ROWS (TH + KH_ - 1)        // 12
#define COLS (TW + KW_ - 1)        // 68
#define PLANE (ROWS * COLS)        // 816 floats per ci
#define TILE_TOT (CIN_ * PLANE)    // 3264 floats = 13056 B LDS
#define NTHREADS 256

// Tiny helper kernel: transpose weights [co][tap] -> [tap][co] so the 16
// co-values of a tap are 64B-contiguous (merged s_load_b512, SGPR operands).
__global__ __launch_bounds__(256)
void Dilation2D_transpose_w(const float* __restrict__ w, float* __restrict__ wT) {
    int i = blockIdx.x * blockDim.x + threadIdx.x;
    if (i < COUT_ * NTAPS) {
        int co  = i / NTAPS;
        int tap = i - co * NTAPS;
        wT[tap * COUT_ + co] = w[i];
    }
}

template <bool TRW>
__global__ __launch_bounds__(NTHREADS)
void Dilation2D_kernel(const float* __restrict__ x,
                       const float* __restrict__ w,   // TRW ? [tap][co] : [co][tap]
                       float* __restrict__ out)
{
    __shared__ float s_tile[TILE_TOT];   // layout: [ci][row(12)][col(68)] linear

    const int tid = threadIdx.x;
    const int w0  = blockIdx.x * TW;
    const int h0  = blockIdx.y * TH;
    const int n   = blockIdx.z;

    // ---- Phase 1: prefill tile with PAD (covers the out-of-image halo) ----
    for (int idx = tid; idx < TILE_TOT; idx += NTHREADS) s_tile[idx] = PADV;
    __syncthreads();

    // ---- Phase 2: gfx1250 async global->LDS copy of in-bounds elements ----
    // Flat LDS addresses live in the low 32 bits of the generic pointer.
    const uint32_t lds_base = (uint32_t)(uintptr_t)&s_tile[0];
    for (int idx = tid; idx < TILE_TOT; idx += NTHREADS) {
        int ci  = idx / PLANE;
        int rem = idx - ci * PLANE;
        int r   = rem / COLS;
        int c   = rem - r * COLS;
        int gh  = h0 - OHA + r;
        int gw  = w0 - OWA + c;
        if ((unsigned)gh < (unsigned)H_ && (unsigned)gw < (unsigned)W_) {
            uint32_t goff  = (uint32_t)((((n * CIN_ + ci) * H_ + gh) * W_ + gw) * 4);
            uint32_t laddr = lds_base + (uint32_t)idx * 4u;
            // GVS mode: mem = SADDR(64b sgpr pair) + VADDR(32b offset); dest LDS
            // byte address in the VDST VGPR. Tracked with ASYNCcnt.
            asm volatile("global_load_async_to_lds_b32 %0, %1, %2 offset:0"
                         :
                         : "v"(laddr), "v"(goff), "s"(x)
                         : "memory");
        }
    }
    asm volatile("s_wait_asynccnt 0" ::: "memory");
    __syncthreads();

    // ---- Phase 3: max-plus accumulation ----
    // thread -> (col wl in [0,64), row r0 in [0,4)); handles rows r0 and r0+4.
    const int wl = tid & (TW - 1);
    const int r0 = tid >> 6;

    float a0[COUT_], a1[COUT_];
#pragma unroll
    for (int co = 0; co < COUT_; ++co) { a0[co] = PADV; a1[co] = PADV; }

#pragma unroll 1
    for (int ci = 0; ci < CIN_; ++ci) {
#pragma unroll 1
        for (int kh = 0; kh < KH_; ++kh) {
            const float* row0 = &s_tile[ci * PLANE + (r0 + kh) * COLS + wl];
            const float* row1 = row0 + 4 * COLS;
            float x0[KW_], x1[KW_];
#pragma unroll
            for (int kw = 0; kw < KW_; ++kw) { x0[kw] = row0[kw]; x1[kw] = row1[kw]; }

            const int tap0 = (ci * KH_ + kh) * KW_;
#pragma unroll
            for (int co = 0; co < COUT_; ++co) {
                float m0 = a0[co];
                float m1 = a1[co];
#pragma unroll
                for (int kw = 0; kw < KW_; ++kw) {
                    // Uniform address -> s_load; SGPR broadcast operand of v_add.
                    float wv = TRW ? w[(tap0 + kw) * COUT_ + co]
                                   : w[co * NTAPS + tap0 + kw];
                    m0 = fmaxf(m0, x0[kw] + wv);   // max chains fold to v_max3_num_f32
                    m1 = fmaxf(m1, x1[kw] + wv);
                }
                a0[co] = m0;
                a1[co] = m1;
            }
        }
    }

    // ---- Phase 4: streaming (non-temporal) stores of the 128MB output ----
    const int gw  = w0 + wl;
    const int gh0 = h0 + r0;
    const int gh1 = gh0 + 4;
#pragma unroll
    for (int co = 0; co < COUT_; ++co) {
        size_t base = (size_t)(n * COUT_ + co) * (size_t)H_;
        __builtin_nontemporal_store(a0[co], &out[(base + gh0) * W_ + gw]);
        __builtin_nontemporal_store(a1[co], &out[(base + gh1) * W_ + gw]);
    }
}

extern "C" void kernel_launch(void* const* d_in, const int* in_sizes, int n_in,
                              void* d_out, int out_size, void* d_ws, size_t ws_size,
                              hipStream_t stream) {
    const float* x = (const float*)d_in[0];
    const float* w = (const float*)d_in[1];
    float* out     = (float*)d_out;

    dim3 grid(W_ / TW, H_ / TH, N_);   // 8 x 64 x 8 workgroups

    const size_t wbytes = (size_t)COUT_ * NTAPS * sizeof(float);  // 6400 B
    if (d_ws != nullptr && ws_size >= wbytes) {
        float* wT = (float*)d_ws;
        Dilation2D_transpose_w<<<(COUT_ * NTAPS + 255) / 256, 256, 0, stream>>>(w, wT);
        Dilation2D_kernel<true><<<grid, NTHREADS, 0, stream>>>(x, wT, out);
    } else {
        Dilation2D_kernel<false><<<grid, NTHREADS, 0, stream>>>(x, w, out);
    }
}